// MutualInformation_JunwenBai_73710228734855
// MI455X (gfx1250) — compile-verified
//
#include <hip/hip_runtime.h>
#include <math.h>

// Problem constants (match reference setup_inputs)
#define BB 128      // batch
#define TT 32       // time steps
#define DF 256      // f feature dim
#define DZ 128      // z feature dim
#define LOG2PI 1.8378770664093453f

typedef __attribute__((ext_vector_type(2))) float v2f;
typedef __attribute__((ext_vector_type(8))) float v8f;

// D = A(16x4 f32) * B(4x16 f32) + C, wave32 WMMA.
// A frag: lane%16 = M row, vgpr r -> K = 2*(lane/16)+r
// B frag: lane%16 = N row (of B^T), vgpr r -> K = 2*(lane/16)+r
// C/D   : vgpr r -> M = r + 8*(lane/16), N = lane%16
__device__ __forceinline__ v8f wmma_k4(v2f a, v2f b, v8f c) {
  return __builtin_amdgcn_wmma_f32_16x16x4_f32(false, a, false, b, (short)0, c,
                                               false, false);
}

// ---------------------------------------------------------------------------
// Stage 1a: f-side weights.  w = exp(-2 lv), a = -2 m w,
// c_j = sum_d (m^2 w + 2 lv) + DF*log(2pi)
__global__ __launch_bounds__(256) void mi_prep_f(const float* __restrict__ fm,
                                                 const float* __restrict__ flv,
                                                 float* __restrict__ fw,
                                                 float* __restrict__ fa,
                                                 float* __restrict__ fc) {
  __shared__ float red[256];
  const int j = blockIdx.x;     // mean row
  const int d = threadIdx.x;    // feature (DF == 256 == blockDim)
  const int idx = j * DF + d;
  const float m = fm[idx];
  const float lv = flv[idx];
  const float w = expf(-2.0f * lv);
  fw[idx] = w;
  fa[idx] = -2.0f * m * w;
  red[d] = m * m * w + 2.0f * lv;
  __syncthreads();
  for (int s = 128; s > 0; s >>= 1) {
    if (d < s) red[d] += red[d + s];
    __syncthreads();
  }
  if (d == 0) fc[j] = red[0] + (float)DF * LOG2PI;
}

// Stage 1b: z-side weights, transposed [B,T,DZ] -> [T,B,DZ] layout.
__global__ __launch_bounds__(128) void mi_prep_z(const float* __restrict__ zm,
                                                 const float* __restrict__ zlv,
                                                 float* __restrict__ zw,
                                                 float* __restrict__ za,
                                                 float* __restrict__ zc) {
  __shared__ float red[128];
  const int t = blockIdx.x / BB;
  const int b = blockIdx.x % BB;
  const int d = threadIdx.x;    // DZ == 128 == blockDim
  const int iidx = (b * TT + t) * DZ + d;
  const int oidx = (t * BB + b) * DZ + d;
  const float m = zm[iidx];
  const float lv = zlv[iidx];
  const float w = expf(-2.0f * lv);
  zw[oidx] = w;
  za[oidx] = -2.0f * m * w;
  red[d] = m * m * w + 2.0f * lv;
  __syncthreads();
  for (int s = 64; s > 0; s >>= 1) {
    if (d < s) red[d] += red[d + s];
    __syncthreads();
  }
  if (d == 0) zc[t * BB + b] = red[0] + (float)DZ * LOG2PI;
}

// ---------------------------------------------------------------------------
// Stage 2: F[i,j] = logq_f_sum via WMMA.  8 blocks (i-tiles) x 8 waves (j-tiles)
__global__ __launch_bounds__(256) void mi_gemm_f(const float* __restrict__ fs,
                                                 const float* __restrict__ fw,
                                                 const float* __restrict__ fa,
                                                 const float* __restrict__ fc,
                                                 float* __restrict__ F) {
  const int tid = threadIdx.x;
  const int wv = tid >> 5, lane = tid & 31;
  const int half = lane >> 4, l16 = lane & 15;
  const int i0 = blockIdx.x * 16;
  const int j0 = wv * 16;
  const float* srow = fs + (i0 + l16) * DF + 2 * half;
  const float* wrow = fw + (j0 + l16) * DF + 2 * half;
  const float* arow = fa + (j0 + l16) * DF + 2 * half;
  v8f acc = {};
  for (int k = 0; k < DF; k += 4) {
    v2f s = {srow[k], srow[k + 1]};
    v2f bw = {wrow[k], wrow[k + 1]};
    v2f ba = {arow[k], arow[k + 1]};
    acc = wmma_k4(s * s, bw, acc);   // sum s^2 * w
    acc = wmma_k4(s, ba, acc);       // sum s * (-2 m w)
  }
  const float c = fc[j0 + l16];
#pragma unroll
  for (int r = 0; r < 8; ++r) {
    const int gi = i0 + r + 8 * half;
    F[gi * BB + j0 + l16] = -0.5f * (acc[r] + c);
  }
}

// Stage 3: logq_f[i] = logsumexp_j F[i,j] - log(B*num_train)
__global__ __launch_bounds__(128) void mi_lse_f(const float* __restrict__ F,
                                                const int* __restrict__ nt,
                                                float* __restrict__ lqf) {
  const int i = threadIdx.x;
  const float ln = logf((float)BB) + logf((float)nt[0]);
  float mx = -INFINITY;
  for (int j = 0; j < BB; ++j) mx = fmaxf(mx, F[i * BB + j]);
  float s = 0.0f;
  for (int j = 0; j < BB; ++j) s += expf(F[i * BB + j] - mx);
  lqf[i] = mx + logf(s) - ln;
}

// ---------------------------------------------------------------------------
// Stage 4: per (t, i-tile): Z tile via WMMA -> LDS -> fused dual logsumexp.
__global__ __launch_bounds__(256) void mi_z_fused(const float* __restrict__ zs,
                                                  const float* __restrict__ zw,
                                                  const float* __restrict__ za,
                                                  const float* __restrict__ zc,
                                                  const float* __restrict__ F,
                                                  const float* __restrict__ lqf,
                                                  const int* __restrict__ nt,
                                                  float* __restrict__ rel) {
  __shared__ float zt[16 * 128];   // Z rows i0..i0+15, all j
  __shared__ float ft[16 * 128];   // F rows i0..i0+15, all j
  const int tid = threadIdx.x;
  const int wv = tid >> 5, lane = tid & 31;
  const int half = lane >> 4, l16 = lane & 15;
  const int i0 = blockIdx.x * 16;
  const int t = blockIdx.y;
#pragma unroll
  for (int e = tid; e < 16 * 128; e += 256)  // uniform 8 iterations, no divergence
    ft[e] = F[(i0 + (e >> 7)) * BB + (e & 127)];
  const int j0 = wv * 16;
  const float* srow = zs + ((size_t)(i0 + l16) * TT + t) * DZ + 2 * half;
  const float* wrow = zw + ((size_t)t * BB + j0 + l16) * DZ + 2 * half;
  const float* arow = za + ((size_t)t * BB + j0 + l16) * DZ + 2 * half;
  v8f acc = {};
  for (int k = 0; k < DZ; k += 4) {
    v2f s = {srow[k], srow[k + 1]};
    v2f bw = {wrow[k], wrow[k + 1]};
    v2f ba = {arow[k], arow[k + 1]};
    acc = wmma_k4(s * s, bw, acc);
    acc = wmma_k4(s, ba, acc);
  }
  const float c = zc[t * BB + j0 + l16];
#pragma unroll
  for (int r = 0; r < 8; ++r)
    zt[(r + 8 * half) * 128 + j0 + l16] = -0.5f * (acc[r] + c);
  __syncthreads();
  if (tid < 16) {  // one thread per i row: fused logsumexp over j for z and f+z
    const float ln = logf((float)BB) + logf((float)nt[0]);
    const float* zr = zt + tid * 128;
    const float* fr = ft + tid * 128;
    float m1 = -INFINITY, m2 = -INFINITY;
    for (int j = 0; j < BB; ++j) {
      const float z = zr[j];
      m1 = fmaxf(m1, z);
      m2 = fmaxf(m2, z + fr[j]);
    }
    float s1 = 0.0f, s2 = 0.0f;
    for (int j = 0; j < BB; ++j) {
      const float z = zr[j];
      s1 += expf(z - m1);
      s2 += expf(z + fr[j] - m2);
    }
    const float lz = m1 + logf(s1) - ln;
    const float lfz = m2 + logf(s2) - ln;
    rel[t * BB + i0 + tid] = fmaxf(lfz - lqf[i0 + tid] - lz, 0.0f);
  }
}

// Stage 5: deterministic mean of rel[T*B]
__global__ __launch_bounds__(256) void mi_final(const float* __restrict__ rel,
                                                float* __restrict__ out) {
  __shared__ float red[256];
  const int tid = threadIdx.x;
  float s = 0.0f;
  for (int e = tid; e < TT * BB; e += 256) s += rel[e];
  red[tid] = s;
  __syncthreads();
  for (int st = 128; st > 0; st >>= 1) {
    if (tid < st) red[tid] += red[tid + st];
    __syncthreads();
  }
  if (tid == 0) out[0] = red[0] / (float)(TT * BB);
}

// ---------------------------------------------------------------------------
extern "C" void kernel_launch(void* const* d_in, const int* in_sizes, int n_in,
                              void* d_out, int out_size, void* d_ws,
                              size_t ws_size, hipStream_t stream) {
  (void)in_sizes; (void)n_in; (void)out_size; (void)ws_size;
  const float* f_mean = (const float*)d_in[0];
  const float* f_logvar = (const float*)d_in[1];
  const float* f_sample = (const float*)d_in[2];
  const float* z_mean = (const float*)d_in[3];
  const float* z_logvar = (const float*)d_in[4];
  const float* z_sample = (const float*)d_in[5];
  const int* num_train = (const int*)d_in[6];
  float* out = (float*)d_out;

  float* ws = (float*)d_ws;                 // ~4.6 MB total
  float* fw = ws;                           // [BB*DF]
  float* fa = fw + BB * DF;                 // [BB*DF]
  float* fc = fa + BB * DF;                 // [BB]
  float* F = fc + BB;                       // [BB*BB]
  float* lqf = F + BB * BB;                 // [BB]
  float* zw = lqf + BB;                     // [TT*BB*DZ]
  float* za = zw + TT * BB * DZ;            // [TT*BB*DZ]
  float* zc = za + TT * BB * DZ;            // [TT*BB]
  float* rel = zc + TT * BB;                // [TT*BB]

  mi_prep_f<<<BB, 256, 0, stream>>>(f_mean, f_logvar, fw, fa, fc);
  mi_prep_z<<<TT * BB, 128, 0, stream>>>(z_mean, z_logvar, zw, za, zc);
  mi_gemm_f<<<BB / 16, 256, 0, stream>>>(f_sample, fw, fa, fc, F);
  mi_lse_f<<<1, BB, 0, stream>>>(F, num_train, lqf);
  mi_z_fused<<<dim3(BB / 16, TT), 256, 0, stream>>>(z_sample, zw, za, zc, F,
                                                    lqf, num_train, rel);
  mi_final<<<1, 256, 0, stream>>>(rel, out);
}